// ConvolutionDownsample_59760174956853
// MI455X (gfx1250) — compile-verified
//
#include <hip/hip_runtime.h>
#include <stdint.h>

typedef __attribute__((ext_vector_type(2))) float v2f;
typedef __attribute__((ext_vector_type(8))) float v8f;

#define B_     4
#define D2_    32
#define CIN_   128
#define COUT_  256
#define NSITE_ (B_*D2_*D2_*D2_)   /* 131072 output sites */
#define LEAK_  0.333f
#define EPS_   1e-4f

/* ---- workspace layout (bytes) ---- */
#define OFF_MASK      0                      /* NSITE_ ints   */
#define OFF_BINCNT    (NSITE_*4)             /* 8 ints        */
#define OFF_BINSTART  (OFF_BINCNT  + 64)     /* 9 ints        */
#define OFF_BINCUR    (OFF_BINSTART+ 64)     /* 8 ints        */
#define OFF_TILEBASE  (OFF_BINCUR  + 64)     /* 9 ints        */
#define OFF_ACTCNT    (OFF_TILEBASE+ 64)     /* 1 int         */
#define OFF_SUM       (OFF_ACTCNT  + 192)    /* 256 floats    */
#define OFF_SUMSQ     (OFF_SUM     + 1024)
#define OFF_SCALE     (OFF_SUMSQ   + 1024)
#define OFF_SHIFT     (OFF_SCALE   + 1024)
#define ZERO_BYTES    (OFF_SHIFT   + 1024)
#define OFF_PERM      ZERO_BYTES             /* n ints, then n ints for sites */

/* ---------- pass 1: occupancy mask + per-corner bin counts ---------- */
__global__ void k_mask_count(const int* __restrict__ coords, const int* __restrict__ bidx,
                             int* __restrict__ mask, int* __restrict__ binCnt, int n) {
    int i = blockIdx.x * blockDim.x + threadIdx.x;
    if (i >= n) return;
    int x = coords[3*i], y = coords[3*i+1], z = coords[3*i+2];
    int b = bidx[i];
    int site = ((b*D2_ + (x>>1))*D2_ + (y>>1))*D2_ + (z>>1);
    mask[site] = 1;
    int c = ((x&1)*2 + (y&1))*2 + (z&1);
    atomicAdd(&binCnt[c], 1);
}

/* ---------- pass 2: tiny 8-bin exclusive scan + tile bases ---------- */
__global__ void k_scan(const int* __restrict__ binCnt, int* __restrict__ binStart,
                       int* __restrict__ binCur, int* __restrict__ tileBase) {
    if (threadIdx.x == 0 && blockIdx.x == 0) {
        int s = 0, tb = 0;
        for (int c = 0; c < 8; ++c) {
            binStart[c] = s; binCur[c] = s; tileBase[c] = tb;
            int cnt = binCnt[c];
            s  += cnt;
            tb += (cnt + 15) >> 4;
        }
        binStart[8] = s; tileBase[8] = tb;
    }
}

/* ---------- pass 3: scatter point ids into corner-binned order ---------- */
__global__ void k_scatter(const int* __restrict__ coords, const int* __restrict__ bidx,
                          int* __restrict__ binCur, int* __restrict__ perm,
                          int* __restrict__ siteArr, int n) {
    int i = blockIdx.x * blockDim.x + threadIdx.x;
    if (i >= n) return;
    int x = coords[3*i], y = coords[3*i+1], z = coords[3*i+2];
    int b = bidx[i];
    int site = ((b*D2_ + (x>>1))*D2_ + (y>>1))*D2_ + (z>>1);
    int c = ((x&1)*2 + (y&1))*2 + (z&1);
    int pos = atomicAdd(&binCur[c], 1);
    perm[pos]    = i;
    siteArr[pos] = site;
}

/* ---------- pass 4: fp32 WMMA GEMM (16pts x 256cols x K=128) + atomic scatter ----------
 * One block = one 16-point tile of one corner bin. 4 waves; wave w owns cols [64w,64w+64)
 * as four 16x16 WMMA tiles sharing one A fragment per K-step.
 * A (16x4 f32, ISA 7.12.2): lane L<16 -> M=L,K={0,1}; lane>=16 -> M=L-16,K={2,3}.
 * B (4x16):  VGPR0: K=0(lo lanes)/K=2(hi lanes), VGPR1: K=1/K=3, N=lane&15.
 * C (16x16): VGPR r: M=r(lo)/r+8(hi), N=lane&15.
 * Tail rows load a clamped (valid) feature row unconditionally: garbage in A row m only
 * affects C row m, which is never scattered (guarded by g < rowEnd).                    */
__global__ void __launch_bounds__(128)
k_gemm(const float* __restrict__ feat, const float* __restrict__ Wt,
       const int* __restrict__ binStart, const int* __restrict__ tileBase,
       const int* __restrict__ perm, const int* __restrict__ siteArr,
       float* __restrict__ out) {
    int t = blockIdx.x;
    if (t >= tileBase[8]) return;
    int c = 0;
#pragma unroll
    for (int i = 0; i < 8; ++i) if (t >= tileBase[i+1]) c = i + 1;
    int mtile   = t - tileBase[c];
    int rowBase = binStart[c] + mtile * 16;
    int rowEnd  = binStart[c+1];

    int tid  = threadIdx.x;
    int lane = tid & 31;
    int wave = tid >> 5;           /* 4 waves */
    int hi   = lane >> 4;          /* 0 = lanes 0-15, 1 = lanes 16-31 */
    int lan  = lane & 15;
    int koff = hi * 2;

    int gA   = rowBase + lan;
    int gCl  = gA < rowEnd ? gA : rowBase;   /* rowBase always valid for a live tile */
    int frow = perm[gCl];
    const float* fp = feat + (size_t)frow * CIN_ + koff;

    const float* wc = Wt + (size_t)c * CIN_ * COUT_;
    int  n0         = wave * 64 + lan;
    const float* wb = wc + (size_t)koff * COUT_ + n0;

    v8f acc0 = {}; v8f acc1 = {}; v8f acc2 = {}; v8f acc3 = {};
    for (int k = 0; k < CIN_; k += 4) {
        v2f a = *(const v2f*)(fp + k);                      /* unconditional A load */
        __builtin_prefetch(fp + k + 16, 0, 3);              /* near-scope prefetch  */
        size_t o = (size_t)k * COUT_;
        v2f b0, b1, b2, b3;
        b0.x = wb[o];      b0.y = wb[o + COUT_];
        b1.x = wb[o + 16]; b1.y = wb[o + COUT_ + 16];
        b2.x = wb[o + 32]; b2.y = wb[o + COUT_ + 32];
        b3.x = wb[o + 48]; b3.y = wb[o + COUT_ + 48];
        acc0 = __builtin_amdgcn_wmma_f32_16x16x4_f32(false, a, false, b0,
                                                     (short)0, acc0, false, false);
        acc1 = __builtin_amdgcn_wmma_f32_16x16x4_f32(false, a, false, b1,
                                                     (short)0, acc1, false, false);
        acc2 = __builtin_amdgcn_wmma_f32_16x16x4_f32(false, a, false, b2,
                                                     (short)0, acc2, false, false);
        acc3 = __builtin_amdgcn_wmma_f32_16x16x4_f32(false, a, false, b3,
                                                     (short)0, acc3, false, false);
    }

#pragma unroll
    for (int r = 0; r < 8; ++r) {
        int M = r + hi * 8;
        int g = rowBase + M;
        if (g < rowEnd) {
            int s = siteArr[g];
            float* po = out + (size_t)s * COUT_ + n0;
            atomicAdd(po,      acc0[r]);
            atomicAdd(po + 16, acc1[r]);
            atomicAdd(po + 32, acc2[r]);
            atomicAdd(po + 48, acc3[r]);
        }
    }
}

/* ---------- pass 5: bias at active sites + per-channel sum/sumsq + active count ---------- */
__global__ void __launch_bounds__(256)
k_stats(const int* __restrict__ mask, const float* __restrict__ bias,
        float* __restrict__ out, float* __restrict__ sum, float* __restrict__ sumsq,
        int* __restrict__ actCnt) {
    int t = threadIdx.x;                 /* channel */
    int siteBase = blockIdx.x * 256;
    float bt = bias[t];
    float s = 0.f, s2 = 0.f;
    int cntLocal = 0;
    for (int j = 0; j < 256; ++j) {
        int site = siteBase + j;
        if (mask[site]) {
            size_t off = (size_t)site * COUT_ + t;
            float v = out[off] + bt;
            out[off] = v;
            s += v; s2 += v * v;
            cntLocal += 1;
        }
    }
    atomicAdd(&sum[t],   s);
    atomicAdd(&sumsq[t], s2);
    if (t == 0) atomicAdd(actCnt, cntLocal);
}

/* ---------- pass 6: fold BN stats into per-channel scale/shift ---------- */
__global__ void k_final(const float* __restrict__ sum, const float* __restrict__ sumsq,
                        const int* __restrict__ actCnt, const float* __restrict__ gamma,
                        const float* __restrict__ beta, float* __restrict__ scale,
                        float* __restrict__ shift) {
    int t = threadIdx.x;
    float cnt  = (float)(*actCnt);
    float mean = sum[t] / cnt;
    float var  = sumsq[t] / cnt - mean * mean;
    float sc   = rsqrtf(var + EPS_) * gamma[t];
    scale[t] = sc;
    shift[t] = beta[t] - mean * sc;
}

/* ---------- pass 7: fused BN + LeakyReLU + mask, float4 vectorized ---------- */
__global__ void __launch_bounds__(256)
k_bnact(const int* __restrict__ mask, const float* __restrict__ scale,
        const float* __restrict__ shift, float* __restrict__ out) {
    size_t i    = (size_t)blockIdx.x * 256 + threadIdx.x;   /* float4 index */
    size_t site = i >> 6;                                   /* 64 float4 per site */
    if (!mask[site]) return;                                /* inactive rows stay zero */
    int cv = (int)(i & 63);
    float4 v  = ((float4*)out)[i];
    float4 sc = ((const float4*)scale)[cv];
    float4 sh = ((const float4*)shift)[cv];
    float4 y;
    y.x = v.x * sc.x + sh.x;  y.x = y.x >= 0.f ? y.x : LEAK_ * y.x;
    y.y = v.y * sc.y + sh.y;  y.y = y.y >= 0.f ? y.y : LEAK_ * y.y;
    y.z = v.z * sc.z + sh.z;  y.z = y.z >= 0.f ? y.z : LEAK_ * y.z;
    y.w = v.w * sc.w + sh.w;  y.w = y.w >= 0.f ? y.w : LEAK_ * y.w;
    ((float4*)out)[i] = y;
}

extern "C" void kernel_launch(void* const* d_in, const int* in_sizes, int n_in,
                              void* d_out, int out_size, void* d_ws, size_t ws_size,
                              hipStream_t stream) {
    const float* feat  = (const float*)d_in[0];
    const int*   coords= (const int*)  d_in[1];
    const int*   bidx  = (const int*)  d_in[2];
    const float* Wt    = (const float*)d_in[3];
    const float* bias  = (const float*)d_in[4];
    const float* gamma = (const float*)d_in[5];
    const float* beta  = (const float*)d_in[6];
    float* out = (float*)d_out;
    char*  ws  = (char*)d_ws;
    const int n = in_sizes[2];           /* number of points */

    int*   mask     = (int*)  (ws + OFF_MASK);
    int*   binCnt   = (int*)  (ws + OFF_BINCNT);
    int*   binStart = (int*)  (ws + OFF_BINSTART);
    int*   binCur   = (int*)  (ws + OFF_BINCUR);
    int*   tileBase = (int*)  (ws + OFF_TILEBASE);
    int*   actCnt   = (int*)  (ws + OFF_ACTCNT);
    float* sum      = (float*)(ws + OFF_SUM);
    float* sumsq    = (float*)(ws + OFF_SUMSQ);
    float* scale    = (float*)(ws + OFF_SCALE);
    float* shift    = (float*)(ws + OFF_SHIFT);
    int*   perm     = (int*)  (ws + OFF_PERM);
    int*   siteArr  = perm + n;

    /* zero output accumulator + mask/counters/stats */
    hipMemsetAsync(out, 0, (size_t)NSITE_ * COUT_ * sizeof(float), stream);
    hipMemsetAsync(ws, 0, ZERO_BYTES, stream);

    int pb = (n + 255) / 256;
    k_mask_count<<<pb, 256, 0, stream>>>(coords, bidx, mask, binCnt, n);
    k_scan      <<<1, 32, 0, stream>>>(binCnt, binStart, binCur, tileBase);
    k_scatter   <<<pb, 256, 0, stream>>>(coords, bidx, binCur, perm, siteArr, n);

    int maxTiles = (n + 15) / 16 + 8;    /* upper bound incl. per-bin padding */
    k_gemm<<<maxTiles, 128, 0, stream>>>(feat, Wt, binStart, tileBase, perm, siteArr, out);

    k_stats<<<NSITE_ / 256, 256, 0, stream>>>(mask, bias, out, sum, sumsq, actCnt);
    k_final<<<1, 256, 0, stream>>>(sum, sumsq, actCnt, gamma, beta, scale, shift);

    int nQuads = NSITE_ * COUT_ / 4;     /* 8388608 float4 */
    k_bnact<<<nQuads / 256, 256, 0, stream>>>(mask, scale, shift, out);
}